// DGL_ChebConv_82772609728706
// MI455X (gfx1250) — compile-verified
//
#include <hip/hip_runtime.h>

typedef __attribute__((ext_vector_type(2))) float v2f;
typedef __attribute__((ext_vector_type(8))) float v8f;

#define D_FEAT 64
constexpr float LAM = 2.0f;

// ---------------- utility kernels ----------------

__global__ void zero_kernel(float* __restrict__ p, long n) {
  long i = (long)blockIdx.x * blockDim.x + threadIdx.x;
  long stride = (long)gridDim.x * blockDim.x;
  for (; i < n; i += stride) p[i] = 0.0f;
}

__global__ void deg_kernel(const int* __restrict__ dst, float* __restrict__ deg, int E) {
  int i = blockIdx.x * blockDim.x + threadIdx.x;
  if (i < E) atomicAdd(&deg[dst[i]], 1.0f);
}

__global__ void norm_kernel(float* __restrict__ d, int N) {
  int i = blockIdx.x * blockDim.x + threadIdx.x;
  if (i < N) {
    float v = fmaxf(d[i], 1.0f);
    d[i] = rsqrtf(v);
  }
}

// One wave (32 lanes) per edge; each lane moves 2 consecutive floats (b64 load + 2 atomics).
__global__ void scatter_kernel(const float* __restrict__ x, const float* __restrict__ nrm,
                               const int* __restrict__ src, const int* __restrict__ dst,
                               float* __restrict__ agg, int E) {
  int lane = threadIdx.x & 31;
  int edge = blockIdx.x * (blockDim.x >> 5) + (threadIdx.x >> 5);
  if (edge >= E) return;
  int s = src[edge];
  int d = dst[edge];
  float ns = nrm[s];
  const float2* xs = (const float2*)(x + (long)s * D_FEAT);
  float2 v = xs[lane];
  float* o = agg + (long)d * D_FEAT + lane * 2;
  atomicAdd(o + 0, v.x * ns);
  atomicAdd(o + 1, v.y * ns);
}

// out[i] = cH * (agg[i]*norm[node]) + cX * t1[i] + cP * t2[i]
// (out may alias agg/t1/t2: each thread reads and writes only its own element)
__global__ void combine_kernel(const float* __restrict__ agg, const float* __restrict__ nrm,
                               const float* __restrict__ t1, const float* __restrict__ t2,
                               float* __restrict__ out,
                               float cH, float cX, float cP, long n) {
  long i = (long)blockIdx.x * blockDim.x + threadIdx.x;
  if (i >= n) return;
  long node = i >> 6;  // D_FEAT == 64
  float h = agg[i] * nrm[node];
  out[i] = cH * h + cX * t1[i] + cP * t2[i];
}

// ---------------- fused WMMA GEMM over all 4 Chebyshev terms ----------------
// out[16 rows x 64 cols] = bias + sum_t Tx_t @ W_t.
// Block = 128 threads = 4 waves; wave w owns output columns [w*16, w*16+16).
// Per term: 16 chained f32 WMMAs (K chunks of 4) -> 64 WMMAs per wave total.

__device__ __forceinline__ v8f accum_term(v8f acc, const float* __restrict__ xr,
                                          const float* __restrict__ Wt,
                                          int khalf, int ncol) {
  // B layout: VGPR0 = W[kchunk*4 + khalf*2 + 0][ncol], VGPR1 = next K row.
  v2f bfrag[16];
#pragma unroll
  for (int kk = 0; kk < 16; ++kk) {
    int krow = kk * 4 + khalf * 2;
    v2f bf;
    bf.x = Wt[(krow + 0) * D_FEAT + ncol];
    bf.y = Wt[(krow + 1) * D_FEAT + ncol];
    bfrag[kk] = bf;
  }
#pragma unroll
  for (int kk = 0; kk < 16; ++kk) {
    v2f a;
    a.x = xr[kk * 4 + khalf * 2 + 0];
    a.y = xr[kk * 4 + khalf * 2 + 1];
    acc = __builtin_amdgcn_wmma_f32_16x16x4_f32(
        /*neg_a=*/false, a, /*neg_b=*/false, bfrag[kk],
        /*c_mod=*/(short)0, acc, /*reuse_a=*/false, /*reuse_b=*/false);
  }
  return acc;
}

__global__ __launch_bounds__(128) void gemm_fused_kernel(
    const float* __restrict__ X0, const float* __restrict__ X1,
    const float* __restrict__ X2, const float* __restrict__ X3,
    const float* __restrict__ W, const float* __restrict__ bias,
    float* __restrict__ out, int N) {
  int lane = threadIdx.x & 31;
  int wv   = threadIdx.x >> 5;   // 0..3 -> output column tile
  int row0 = blockIdx.x * 16;
  if (row0 >= N) return;

  int mlo   = lane & 15;         // A row within tile / B,C,D column within tile
  int khalf = lane >> 4;         // 0: K={0,1}; 1: K={2,3} within each chunk
  int ncol  = wv * 16 + mlo;     // global output column (0..63)

  int arow = row0 + mlo;
  if (arow >= N) arow = N - 1;   // tail clamp (N%16==0 here, kept for generality)
  long aoff = (long)arow * D_FEAT;

  v8f acc;
  {
    float bb = bias[ncol];
#pragma unroll
    for (int v = 0; v < 8; ++v) acc[v] = bb;
  }

  acc = accum_term(acc, X0 + aoff, W + 0 * D_FEAT * D_FEAT, khalf, ncol);
  acc = accum_term(acc, X1 + aoff, W + 1 * D_FEAT * D_FEAT, khalf, ncol);
  acc = accum_term(acc, X2 + aoff, W + 2 * D_FEAT * D_FEAT, khalf, ncol);
  acc = accum_term(acc, X3 + aoff, W + 3 * D_FEAT * D_FEAT, khalf, ncol);

  // C/D layout: VGPR v holds M = v + 8*khalf, N = ncol.
  if (row0 + 16 <= N) {
    float* orow = out + (long)(row0 + khalf * 8) * D_FEAT + ncol;
#pragma unroll
    for (int v = 0; v < 8; ++v) orow[(long)v * D_FEAT] = acc[v];
  } else {
#pragma unroll
    for (int v = 0; v < 8; ++v) {
      int r = row0 + khalf * 8 + v;
      if (r < N) out[(long)r * D_FEAT + ncol] = acc[v];
    }
  }
}

// ---------------- launch ----------------

extern "C" void kernel_launch(void* const* d_in, const int* in_sizes, int n_in,
                              void* d_out, int out_size, void* d_ws, size_t ws_size,
                              hipStream_t stream) {
  const float* feat = (const float*)d_in[0];   // (N, 64)  == Tx0
  const int*   src  = (const int*)d_in[1];     // (E,)
  const int*   dst  = (const int*)d_in[2];     // (E,)
  const float* W    = (const float*)d_in[3];   // (4, 64, 64)
  const float* b    = (const float*)d_in[4];   // (64,)
  float*       out  = (float*)d_out;           // (N, 64)

  const int N = in_sizes[0] / D_FEAT;
  const int E = in_sizes[1];
  const long nd = (long)N * D_FEAT;

  float* nrm = (float*)d_ws;                   // N floats (deg, then norm in-place)
  float* agg = nrm + N;                        // N*64 scratch; finally holds Tx3
  float* txA = agg + nd;                       // N*64 -> Tx1
  float* txB = txA + nd;                       // N*64 -> Tx2

  const int cgrid = (int)((nd + 255) / 256);
  const int sgrid = (E + 7) / 8;               // 8 edges per 256-thread block
  const int ggrid = (N + 15) / 16;

  // degree -> norm = rsqrt(max(deg,1))
  zero_kernel<<<512, 256, 0, stream>>>(nrm, (long)N);
  deg_kernel<<<(E + 255) / 256, 256, 0, stream>>>(dst, nrm, E);
  norm_kernel<<<(N + 255) / 256, 256, 0, stream>>>(nrm, N);

  // Tx1 = -(2/lam)*prop(Tx0) + (2/lam - 1)*Tx0
  zero_kernel<<<2048, 256, 0, stream>>>(agg, nd);
  scatter_kernel<<<sgrid, 256, 0, stream>>>(feat, nrm, src, dst, agg, E);
  combine_kernel<<<cgrid, 256, 0, stream>>>(agg, nrm, feat, feat, txA,
                                            -2.0f / LAM, 2.0f / LAM - 1.0f, 0.0f, nd);

  // Tx2 = -(4/lam)*prop(Tx1) + (4/lam - 2)*Tx1 - Tx0
  zero_kernel<<<2048, 256, 0, stream>>>(agg, nd);
  scatter_kernel<<<sgrid, 256, 0, stream>>>(txA, nrm, src, dst, agg, E);
  combine_kernel<<<cgrid, 256, 0, stream>>>(agg, nrm, txA, feat, txB,
                                            -4.0f / LAM, 4.0f / LAM - 2.0f, -1.0f, nd);

  // Tx3 = -(4/lam)*prop(Tx2) + (4/lam - 2)*Tx2 - Tx1   (written in place into agg)
  zero_kernel<<<2048, 256, 0, stream>>>(agg, nd);
  scatter_kernel<<<sgrid, 256, 0, stream>>>(txB, nrm, src, dst, agg, E);
  combine_kernel<<<cgrid, 256, 0, stream>>>(agg, nrm, txB, txA, agg,
                                            -4.0f / LAM, 4.0f / LAM - 2.0f, -1.0f, nd);

  // out = b + Tx0@W0 + Tx1@W1 + Tx2@W2 + Tx3@W3  (single pass, no RMW)
  gemm_fused_kernel<<<ggrid, 128, 0, stream>>>(feat, txA, txB, agg, W, b, out, N);
}